// GCNSharedD1_55070070669890
// MI455X (gfx1250) — compile-verified
//
#include <hip/hip_runtime.h>

#define B_ 128
#define N_ 50000
#define L_ 8
#define E_ 250000
#define U_ 10000
#define G_ 20000
#define R_ 2000
#define C_ 2

typedef __attribute__((ext_vector_type(2))) float v2f;
typedef __attribute__((ext_vector_type(8))) float v8f;

// ---------------------------------------------------------------------------
// Zero h[N][B] and agg[U][B] (float4-vectorized).
// ---------------------------------------------------------------------------
__global__ void init_zero_kernel(float4* __restrict__ h4, float4* __restrict__ a4) {
    long idx = (long)blockIdx.x * blockDim.x + threadIdx.x;
    const long nh = (long)N_ * B_ / 4;
    const long na = (long)U_ * B_ / 4;
    float4 z; z.x = 0.f; z.y = 0.f; z.z = 0.f; z.w = 0.f;
    if (idx < nh)            h4[idx] = z;
    else if (idx < nh + na)  a4[idx - nh] = z;
}

// ---------------------------------------------------------------------------
// h[gene_map[g]][b] = X[b][g]   (X read coalesced, scattered column writes)
// ---------------------------------------------------------------------------
__global__ void gene_scatter_kernel(const float* __restrict__ X,
                                    const int* __restrict__ gene_map,
                                    float* __restrict__ h) {
    long idx = (long)blockIdx.x * blockDim.x + threadIdx.x;  // = b*G + g
    if (idx >= (long)B_ * G_) return;
    int g = (int)(idx % G_);
    int b = (int)(idx / G_);
    h[(long)gene_map[g] * B_ + b] = X[idx];
}

// ---------------------------------------------------------------------------
// Edge kernel: one wave per edge, each lane handles 4 batch elements (float4).
// agg[dst][b] += h[src][b] * w   via native f32 atomics.
// ---------------------------------------------------------------------------
__global__ void edge_kernel(const float* __restrict__ h,
                            float* __restrict__ agg,
                            const int* __restrict__ src,
                            const int* __restrict__ dst,
                            const float* __restrict__ lw, int li) {
    long idx = (long)blockIdx.x * blockDim.x + threadIdx.x;
    if (idx >= (long)E_ * 32) return;
    int e = (int)(idx >> 5);
    int q = (int)(idx & 31);
    float w = lw[li];
    int s = src[e];
    int d = dst[e];
    const float4* hp = (const float4*)(h + (long)s * B_);
    float4 v = hp[q];
    float* ap = agg + (long)d * B_ + q * 4;
    unsafeAtomicAdd(ap + 0, v.x * w);
    unsafeAtomicAdd(ap + 1, v.y * w);
    unsafeAtomicAdd(ap + 2, v.z * w);
    unsafeAtomicAdd(ap + 3, v.w * w);
}

// ---------------------------------------------------------------------------
// Update kernel: h[dst_unique[u]][b] = tanh(agg[u][b] + bias[dst_unique[u]]);
// also re-zeros agg for the next layer.
// ---------------------------------------------------------------------------
__global__ void update_kernel(float* __restrict__ h,
                              float* __restrict__ agg,
                              const int* __restrict__ dst_unique,
                              const float* __restrict__ bias) {
    long idx = (long)blockIdx.x * blockDim.x + threadIdx.x;
    if (idx >= (long)U_ * B_) return;
    int u = (int)(idx >> 7);
    int b = (int)(idx & 127);
    int nd = dst_unique[u];
    float a = agg[idx];
    agg[idx] = 0.0f;
    h[(long)nd * B_ + b] = tanhf(a + bias[nd]);
}

// ---------------------------------------------------------------------------
// Head: out[b][c] = sum_r h[root_ids[r]][b] * W[c][r] + hb[c]
// One wave per 16-row batch tile; V_WMMA_F32_16X16X4_F32, K=2000 in 500 steps.
// A(16x4 f32): lane m = lane&15; VGPR0 holds K = (lane>=16 ? 2 : 0), VGPR1 = K+1.
// B(4x16 f32): mirrored (lane n = lane&15); columns n>=C are zero-masked.
// C/D(16x16 f32): VGPR v -> M = (lane>>4)*8 + v, N = lane&15.
// ---------------------------------------------------------------------------
__global__ void head_kernel(const float* __restrict__ h,
                            const int* __restrict__ root_ids,
                            const float* __restrict__ W,
                            const float* __restrict__ hb,
                            float* __restrict__ out) {
    int tile = blockIdx.x;        // 0..7 -> batch rows [tile*16, tile*16+16)
    int lane = threadIdx.x;       // 0..31
    int m  = lane & 15;
    int kb = (lane >> 4) * 2;     // 0 or 2
    int mg = tile * 16 + m;       // global batch row for A loads

    int n    = lane & 15;
    int ncl  = (n < C_) ? n : 0;          // clamped column index (avoid OOB)
    float nm = (n < C_) ? 1.0f : 0.0f;    // mask for unused B columns

    v8f acc = {};
    for (int k = 0; k < R_; k += 4) {
        int r0 = root_ids[k + kb];
        int r1 = root_ids[k + kb + 1];
        v2f a, bm;
        a.x = h[(long)r0 * B_ + mg];
        a.y = h[(long)r1 * B_ + mg];
        bm.x = W[ncl * R_ + k + kb]     * nm;
        bm.y = W[ncl * R_ + k + kb + 1] * nm;
        acc = __builtin_amdgcn_wmma_f32_16x16x4_f32(
            /*neg_a=*/false, a, /*neg_b=*/false, bm,
            /*c_mod=*/(short)0, acc, /*reuse_a=*/false, /*reuse_b=*/false);
    }

    int mrow = tile * 16 + ((lane >> 4) * 8);
    if (n < C_) {
        float bias = hb[n];
#pragma unroll
        for (int v = 0; v < 8; ++v) {
            out[(mrow + v) * C_ + n] = acc[v] + bias;
        }
    }
}

// ---------------------------------------------------------------------------
extern "C" void kernel_launch(void* const* d_in, const int* in_sizes, int n_in,
                              void* d_out, int out_size, void* d_ws, size_t ws_size,
                              hipStream_t stream) {
    const float* X          = (const float*)d_in[0];  // (B,G)
    const float* lw         = (const float*)d_in[1];  // (L,)
    const float* bias       = (const float*)d_in[2];  // (N,)
    const float* W          = (const float*)d_in[3];  // (C,R)
    const float* hb         = (const float*)d_in[4];  // (C,)
    const int*   gene_map   = (const int*)d_in[5];    // (G,)
    const int*   src        = (const int*)d_in[6];    // (L,E)
    const int*   dst_pos    = (const int*)d_in[7];    // (L,E)
    const int*   dst_unique = (const int*)d_in[8];    // (L,U)
    const int*   root_ids   = (const int*)d_in[9];    // (R,)
    float* out = (float*)d_out;

    float* h   = (float*)d_ws;                 // [N][B] transposed state
    float* agg = h + (size_t)N_ * B_;          // [U][B]

    {
        long tot4 = ((long)N_ * B_ + (long)U_ * B_) / 4;
        int blocks = (int)((tot4 + 255) / 256);
        init_zero_kernel<<<blocks, 256, 0, stream>>>((float4*)h, (float4*)agg);
    }
    {
        long tot = (long)B_ * G_;
        int blocks = (int)((tot + 255) / 256);
        gene_scatter_kernel<<<blocks, 256, 0, stream>>>(X, gene_map, h);
    }
    for (int li = 0; li < L_; ++li) {
        {
            long tot = (long)E_ * 32;
            int blocks = (int)((tot + 255) / 256);
            edge_kernel<<<blocks, 256, 0, stream>>>(
                h, agg, src + (long)li * E_, dst_pos + (long)li * E_, lw, li);
        }
        {
            long tot = (long)U_ * B_;
            int blocks = (int)((tot + 255) / 256);
            update_kernel<<<blocks, 256, 0, stream>>>(
                h, agg, dst_unique + (long)li * U_, bias);
        }
    }
    head_kernel<<<8, 32, 0, stream>>>(h, root_ids, W, hb, out);
}